// PhysicallyConstrainedRNN_66348654788901
// MI455X (gfx1250) — compile-verified
//
#include <hip/hip_runtime.h>

// MI455X / gfx1250 fused input-projection + ReLU-scan RNN.
// B=32, T=2048, D=512, N=256 (hard-coded from the reference).
//
//   forcing[b,t,n] = u[b,t,:] . W[n,:] + bias[n]          fp32 WMMA 16x16x4
//   h[t] = relu(forcing[t] + h[t-1]*sigmoid(raw_inertia)) register scan
//
// Data movement: TDM (tensor_load_to_lds) DMAs W slice once and double-
// buffered 16x512 u tiles into LDS with hardware padding (2 DWORDs every
// 256 DWORDs -> 516-float row stride, bank-conflict-free ds_load_b64
// fragments). TENSORcnt + workgroup barrier publishes each tile; the next
// tile's DMA overlaps the 128-WMMA tile GEMM.

typedef float v2f  __attribute__((ext_vector_type(2)));
typedef float v8f  __attribute__((ext_vector_type(8)));
typedef unsigned int u32x4 __attribute__((ext_vector_type(4)));
typedef int          i32x4 __attribute__((ext_vector_type(4)));
typedef int          i32x8 __attribute__((ext_vector_type(8)));

#define B_DIM   32
#define T_DIM   2048
#define D_DIM   512
#define N_DIM   256

#define NCOLS   64          // n-columns per block (16 per wave)
#define THREADS 128         // 4 waves (wave32)
#define LSTRIDE 516         // 512 + 2x2 TDM pad DWORDs: rows 8B-aligned,
                            // bank shift 4/row -> frag b64 loads hit all 64 banks

// Issue one TDM 2-D tile load: rows x 512 fp32, row-major contiguous source,
// LDS destination padded +2 DWORDs every 256 DWORDs (row stride 516 floats).
__device__ __forceinline__ void tdm_load_rows(uint32_t lds_byte,
                                              const float* gsrc, int rows)
{
    const uint64_t ga = (uint64_t)(uintptr_t)(const void*)gsrc;

    u32x4 g0;
    g0[0] = 1u;                                   // count=1, user descriptor
    g0[1] = lds_byte;                             // lds_addr (bytes)
    g0[2] = (uint32_t)ga;                         // global_addr[31:0]
    g0[3] = (uint32_t)(ga >> 32) | 0x80000000u;   // global_addr[56:32] | type=2

    i32x8 g1;
    g1[0] = (2 << 16)          // data_size: 4 bytes
          | (1 << 20)          // pad_enable
          | (7 << 22)          // pad_interval: 256 DWORDs
          | (1 << 25);         // pad_amount:   2 DWORDs
    g1[1] = (D_DIM & 0xFFFF) << 16;   // tensor_dim0[15:0] = 512
    g1[2] = rows << 16;               // tensor_dim0[31:16]=0 | tensor_dim1[15:0]
    g1[3] = D_DIM << 16;              // tensor_dim1[31:16]=0 | tile_dim0 = 512
    g1[4] = rows;                     // tile_dim1 | tile_dim2 = 0
    g1[5] = D_DIM;                    // tensor_dim0_stride[31:0] = 512 (contiguous)
    g1[6] = 0;                        // stride[47:32] | tensor_dim1_stride[15:0]
    g1[7] = 0;                        // tensor_dim1_stride[47:16]

    i32x4 gz = {0, 0, 0, 0};          // groups 2/3: 2-D tile, unused
#if defined(__clang_major__) && (__clang_major__ >= 23)
    i32x8 gz8 = {0, 0, 0, 0, 0, 0, 0, 0};
    __builtin_amdgcn_tensor_load_to_lds(g0, g1, gz, gz, gz8, 0);
#else
    __builtin_amdgcn_tensor_load_to_lds(g0, g1, gz, gz, 0);
#endif
}

__global__ __launch_bounds__(THREADS)
void fused_rnn_wmma_tdm(const float* __restrict__ u_seq,
                        const float* __restrict__ W_in,
                        const float* __restrict__ b_in,
                        const float* __restrict__ raw_inertia,
                        float* __restrict__ out)
{
    extern __shared__ float lds[];
    float* w_lds = lds;                         // NCOLS x LSTRIDE
    float* ubuf0 = lds + NCOLS * LSTRIDE;       // 16 x LSTRIDE (double buffered)
    float* ubuf1 = ubuf0 + 16 * LSTRIDE;

    const int tid  = threadIdx.x;
    const int lane = tid & 31;
    const int wave = tid >> 5;                  // 0..3
    const int col  = lane & 15;
    const int half = lane >> 4;                 // 0: K+{0,1}, 1: K+{2,3}
    const int b    = blockIdx.y;
    const int n0g  = blockIdx.x * NCOLS;

    const float* ubase = u_seq + (size_t)b * T_DIM * D_DIM;
    float*       obase = out   + (size_t)b * T_DIM * N_DIM;

    const uint32_t w_off  = (uint32_t)(uintptr_t)(void*)w_lds;
    const uint32_t u_off0 = (uint32_t)(uintptr_t)(void*)ubuf0;
    const uint32_t u_off1 = (uint32_t)(uintptr_t)(void*)ubuf1;

    // Wave 0 kicks off the W-slice DMA and the first u tile (TENSORcnt=2).
    if (wave == 0) {
        tdm_load_rows(w_off, W_in + (size_t)n0g * D_DIM, NCOLS);
        tdm_load_rows(u_off0, ubase, 16);
    }

    // per-lane scan constants (both lane-halves mirror the same column)
    const int   n_glob = n0g + wave * 16 + col;
    const float bias   = b_in[n_glob];
    const float ret    = 1.0f / (1.0f + __expf(-raw_inertia[n_glob]));

    const float* bptr = w_lds + (wave * 16 + col) * LSTRIDE + 2 * half;
    const int    aoff = col * LSTRIDE + 2 * half;

    float h = 0.0f;

    for (int t0 = 0; t0 < T_DIM; t0 += 16) {
        const int ibuf = (t0 >> 4) & 1;

        if (wave == 0) __builtin_amdgcn_s_wait_tensorcnt(0);
        __syncthreads();   // current tile (+W) resident; prev-buffer reads drained

        // overlap: DMA the next tile into the other buffer during this GEMM
        if (wave == 0 && (t0 + 16) < T_DIM)
            tdm_load_rows(ibuf ? u_off0 : u_off1,
                          ubase + (size_t)(t0 + 16) * D_DIM, 16);

        const float* aptr = (ibuf ? ubuf1 : ubuf0) + aoff;

        // ---- 16x16 tile GEMM, K=512 via 128 x v_wmma_f32_16x16x4_f32 ----
        // TDM pad inserts +2 DWORDs after the first 256 DWORDs of each row,
        // so K >= 256 lives at lds offset K+2: split the K loop in halves.
        v8f acc0 = {};
        v8f acc1 = {};
        #pragma unroll 4
        for (int k = 0; k < 256; k += 8) {
            v2f a0 = *(const v2f*)(aptr + k);
            v2f b0 = *(const v2f*)(bptr + k);
            acc0 = __builtin_amdgcn_wmma_f32_16x16x4_f32(
                false, a0, false, b0, (short)0, acc0, false, false);
            v2f a1 = *(const v2f*)(aptr + k + 4);
            v2f b1 = *(const v2f*)(bptr + k + 4);
            acc1 = __builtin_amdgcn_wmma_f32_16x16x4_f32(
                false, a1, false, b1, (short)0, acc1, false, false);
        }
        #pragma unroll 4
        for (int k = 256; k < 512; k += 8) {
            v2f a0 = *(const v2f*)(aptr + k + 2);
            v2f b0 = *(const v2f*)(bptr + k + 2);
            acc0 = __builtin_amdgcn_wmma_f32_16x16x4_f32(
                false, a0, false, b0, (short)0, acc0, false, false);
            v2f a1 = *(const v2f*)(aptr + k + 6);
            v2f b1 = *(const v2f*)(bptr + k + 6);
            acc1 = __builtin_amdgcn_wmma_f32_16x16x4_f32(
                false, a1, false, b1, (short)0, acc1, false, false);
        }

        // ---- gather this column's 16 forcing values into every lane ----
        // D layout: VGPR r, lanes 0-15: (M=r, N=lane); lanes 16-31: (M=r+8, N=lane-16)
        float f[16];
        #pragma unroll
        for (int r = 0; r < 8; ++r) {
            float dr = acc0[r] + acc1[r];
            f[r]     = __shfl(dr, col,      32);   // row t0+r,   col
            f[r + 8] = __shfl(dr, col | 16, 32);   // row t0+r+8, col
        }

        // ---- serial recurrence over the 16 timesteps (registers only) ----
        #pragma unroll
        for (int r = 0; r < 16; ++r) {
            float hn = f[r] + bias + h * ret;
            h = hn > 0.0f ? hn : 0.0f;
            f[r] = h;
        }

        // ---- store in D-layout: two contiguous 64B row segments per store ----
        float* orow = obase + (size_t)(t0 + half * 8) * N_DIM + n_glob;
        #pragma unroll
        for (int r = 0; r < 8; ++r) {
            orow[(size_t)r * N_DIM] = half ? f[r + 8] : f[r];
        }
    }
}

extern "C" void kernel_launch(void* const* d_in, const int* in_sizes, int n_in,
                              void* d_out, int out_size, void* d_ws, size_t ws_size,
                              hipStream_t stream)
{
    (void)in_sizes; (void)n_in; (void)out_size; (void)d_ws; (void)ws_size;
    const float* u_seq       = (const float*)d_in[0];
    const float* W_in        = (const float*)d_in[1];
    const float* b_in        = (const float*)d_in[2];
    const float* raw_inertia = (const float*)d_in[3];
    float* out = (float*)d_out;

    dim3 grid(N_DIM / NCOLS, B_DIM);          // (4, 32)
    dim3 block(THREADS);                      // 4 waves
    size_t smem = (size_t)(NCOLS + 32) * LSTRIDE * sizeof(float);  // ~194 KB of 320 KB
    fused_rnn_wmma_tdm<<<grid, block, smem, stream>>>(u_seq, W_in, b_in,
                                                      raw_inertia, out);
}